// KerasLightGCN_2052994367636
// MI455X (gfx1250) — compile-verified
//
#include <hip/hip_runtime.h>
#include <stdint.h>

#define DIM      64
#define TPB      256
#define CHUNK    1024   // edges staged per LDS buffer (= TPB * 4)

// 12 KB per buffer, double buffered = 24 KB LDS
struct __align__(16) EdgeBuf {
  int   rows[CHUNK];
  int   cols[CHUNK];
  float vals[CHUNK];
};

// raw 32-bit LDS byte address (addrspacecast generic->AS3, then ptrtoint)
__device__ __forceinline__ uint32_t lds_off(const void* p) {
  return (uint32_t)(uintptr_t)(__attribute__((address_space(3))) const void*)p;
}

// gfx1250 async global->LDS copy, 16B per lane, tracked by ASYNCcnt
__device__ __forceinline__ void async_b128(uint32_t lds, const void* g) {
  asm volatile("global_load_async_to_lds_b128 %0, %1, off"
               :: "v"(lds), "v"((uint64_t)(uintptr_t)g) : "memory");
}

// one 256-thread block stages 1024 edges: 3 async B128 per thread
__device__ __forceinline__ void issue_chunk(EdgeBuf* b, int t,
                                            const int* __restrict__ rows,
                                            const int* __restrict__ cols,
                                            const float* __restrict__ vals,
                                            long long base) {
  long long o = base + (long long)t * 4;
  async_b128(lds_off(&b->rows[t * 4]), rows + o);
  async_b128(lds_off(&b->cols[t * 4]), cols + o);
  async_b128(lds_off(&b->vals[t * 4]), vals + o);
}

// 4 consecutive no-return fp32 atomics (one 16B slice of a row)
__device__ __forceinline__ void atomic_add4(float* p, float4 m) {
  asm volatile("global_atomic_add_f32 %0, %1, off\n\t"
               "global_atomic_add_f32 %0, %2, off offset:4\n\t"
               "global_atomic_add_f32 %0, %3, off offset:8\n\t"
               "global_atomic_add_f32 %0, %4, off offset:12"
               :: "v"((uint64_t)(uintptr_t)p),
                  "v"(m.x), "v"(m.y), "v"(m.z), "v"(m.w)
               : "memory");
}

__global__ void __launch_bounds__(TPB)
spmm_scatter(const float* __restrict__ src, float* __restrict__ dst,
             const int* __restrict__ rows, const int* __restrict__ cols,
             const float* __restrict__ vals, int nfull) {
  __shared__ EdgeBuf buf[2];
  const int t    = threadIdx.x;
  const int step = gridDim.x;
  const int g    = t >> 4;   // edge-group within pass (0..15)
  const int l    = t & 15;   // float4 lane within row (0..15)

  int c = blockIdx.x;
  if (c < nfull)
    issue_chunk(&buf[0], t, rows, cols, vals, (long long)c * CHUNK);

  int ib = 0;
  for (; c < nfull; c += step, ib ^= 1) {
    const int nxt = c + step;
    if (nxt < nfull) {
      // prefetch next chunk into the other buffer, then wait for current
      issue_chunk(&buf[ib ^ 1], t, rows, cols, vals, (long long)nxt * CHUNK);
      asm volatile("s_wait_asynccnt 3" ::: "memory");  // oldest 3 (current) done
    } else {
      asm volatile("s_wait_asynccnt 0" ::: "memory");
    }
    __syncthreads();  // current buffer valid for all waves

    EdgeBuf* B = &buf[ib];
    #pragma unroll 4
    for (int p = 0; p < CHUNK / 16; ++p) {     // 64 passes, 16 edges/pass
      const int   e   = p * 16 + g;
      const int   row = B->rows[e];
      const int   col = B->cols[e];
      const float v   = B->vals[e];
      // coalesced 256B row gather: 16 lanes x float4 (L2-resident)
      float4 m = ((const float4*)(src + (size_t)col * DIM))[l];
      m.x *= v; m.y *= v; m.z *= v; m.w *= v;
      atomic_add4(dst + (size_t)row * DIM + l * 4, m);
    }
    __syncthreads();  // everyone done reading buf[ib] before it is refilled
  }
}

// remainder edges (unused for NNZ=3.2M, kept for generality)
__global__ void __launch_bounds__(TPB)
spmm_tail(const float* __restrict__ src, float* __restrict__ dst,
          const int* __restrict__ rows, const int* __restrict__ cols,
          const float* __restrict__ vals, int start, int nnz) {
  const int idx = blockIdx.x * TPB + threadIdx.x;
  const int e   = start + (idx >> 4);
  const int l   = idx & 15;
  if (e < nnz) {
    const int   row = rows[e];
    const int   col = cols[e];
    const float v   = vals[e];
    float4 m = ((const float4*)(src + (size_t)col * DIM))[l];
    m.x *= v; m.y *= v; m.z *= v; m.w *= v;
    atomic_add4(dst + (size_t)row * DIM + l * 4, m);
  }
}

__global__ void __launch_bounds__(TPB)
fill_zero(float4* __restrict__ p, int n4) {
  const int i = blockIdx.x * TPB + threadIdx.x;
  if (i < n4) p[i] = make_float4(0.f, 0.f, 0.f, 0.f);
}

__global__ void __launch_bounds__(TPB)
concat_embed(const float4* __restrict__ u, const float4* __restrict__ it,
             float4* __restrict__ x, int u4, int n4) {
  const int i = blockIdx.x * TPB + threadIdx.x;
  if (i < n4) x[i] = (i < u4) ? u[i] : it[i - u4];
}

extern "C" void kernel_launch(void* const* d_in, const int* in_sizes, int n_in,
                              void* d_out, int out_size, void* d_ws, size_t ws_size,
                              hipStream_t stream) {
  const float* user_emb = (const float*)d_in[0];
  const float* item_emb = (const float*)d_in[1];
  const float* adj_vals = (const float*)d_in[2];
  const int*   adj_row  = (const int*)d_in[3];
  const int*   adj_col  = (const int*)d_in[4];

  const int userElems  = in_sizes[0];            // 50000*64
  const int itemElems  = in_sizes[1];            // 50000*64
  const int nnz        = in_sizes[2];            // 3.2M
  const int totalElems = userElems + itemElems;  // == out_size
  const int n4  = totalElems / 4;
  const int u4  = userElems / 4;

  float* X   = (float*)d_ws;                     // ping
  float* Y   = X + (size_t)totalElems;           // pong
  float* out = (float*)d_out;

  const int cpyBlocks = (n4 + TPB - 1) / TPB;
  const int nfull     = nnz / CHUNK;
  const int rem       = nnz - nfull * CHUNK;
  const int remStart  = nfull * CHUNK;
  int spmmBlocks = nfull < 1024 ? nfull : 1024;

  // X0 = concat(user_emb, item_emb)
  concat_embed<<<cpyBlocks, TPB, 0, stream>>>((const float4*)user_emb,
                                              (const float4*)item_emb,
                                              (float4*)X, u4, n4);

  const float* layer_src[3] = {X, Y, X};
  float*       layer_dst[3] = {Y, X, out};
  for (int layer = 0; layer < 3; ++layer) {
    const float* s = layer_src[layer];
    float*       d = layer_dst[layer];
    fill_zero<<<cpyBlocks, TPB, 0, stream>>>((float4*)d, n4);
    if (spmmBlocks > 0)
      spmm_scatter<<<spmmBlocks, TPB, 0, stream>>>(s, d, adj_row, adj_col,
                                                   adj_vals, nfull);
    if (rem > 0) {
      const int tailThreads = rem * 16;
      spmm_tail<<<(tailThreads + TPB - 1) / TPB, TPB, 0, stream>>>(
          s, d, adj_row, adj_col, adj_vals, remStart, nnz);
    }
  }
}